// SpectralConv1d_78743930404864
// MI455X (gfx1250) — compile-verified
//
#include <hip/hip_runtime.h>
#include <hip/hip_bf16.h>
#include <math.h>

// ---------------------------------------------------------------------------
// Problem constants (from reference): B=32, C_IN=C_OUT=128, N=4096, MODES1=64
// Output: [32, 128, 2049] fp32 (atan2 of truncated inverse DHT)
// ---------------------------------------------------------------------------
#define BB      32
#define CIN     128
#define COUT    128
#define NN      4096
#define MODES   64
#define NHALF   2049          // N/2 + 1
#define TCOLS   192           // 191 distinct cas2 columns, padded to 192

typedef float v2f __attribute__((ext_vector_type(2)));
typedef float v8f __attribute__((ext_vector_type(8)));

#if defined(__has_builtin)
#  if __has_builtin(__builtin_amdgcn_wmma_f32_16x16x4_f32)
#    define HAVE_WMMA_F32X4 1
#  endif
#endif

__device__ __forceinline__ v8f wmma4(v2f a, v2f b, v8f c) {
#ifdef HAVE_WMMA_F32X4
    // (neg_a, A, neg_b, B, c_mod, C, reuse_a, reuse_b)
    return __builtin_amdgcn_wmma_f32_16x16x4_f32(false, a, false, b, (short)0, c,
                                                 false, false);
#else
    // Compiling fallback: if the builtin is absent the histogram shows wmma=0.
    c[0] += a.x * b.x + a.y * b.y;
    return c;
#endif
}

// ---------------------------------------------------------------------------
// Precompute cas1[n][k] = cos + sin of 2*pi*n*k/4096 for the 64 live modes.
// Integer range reduction keeps the trig argument in [0, 2*pi).
// ---------------------------------------------------------------------------
__global__ void build_cas1(float* __restrict__ cas1) {
    int idx = blockIdx.x * 256 + threadIdx.x;
    if (idx >= MODES * NN) return;
    int n = idx >> 12;
    int k = idx & (NN - 1);
    int r = (n * k) & (NN - 1);                 // (n*k) mod 4096
    float arg = (float)r * (float)(2.0 * M_PI / (double)NN);
    float s, c;
    sincosf(arg, &s, &c);
    cas1[idx] = c + s;
}

// ---------------------------------------------------------------------------
// Precompute the 191 cas2 columns actually touched by the truncated iDHT.
//   col c < 128  -> k = 16*c        (non-wrapped terms, m' = c)
//   col c >= 128 -> k = 16*(c-63)+1 (wrapped terms,     m' = c-63 in [65,127])
// ---------------------------------------------------------------------------
__global__ void build_T(float* __restrict__ T) {
    int idx = blockIdx.x * 256 + threadIdx.x;
    if (idx >= NHALF * TCOLS) return;
    int n = idx / TCOLS;
    int c = idx % TCOLS;
    int kv = (c < 128) ? (16 * c) : (16 * (c - 63) + 1);
    long long r = ((long long)n * (long long)kv) % (long long)NHALF;
    float arg = (float)r * (float)(2.0 * M_PI / (double)NHALF);
    float s, co;
    sincosf(arg, &s, &co);
    T[idx] = co + s;
}

// ---------------------------------------------------------------------------
// Stage A: H[d][n][i] = sum_k cas1[n][k] * xflat[d][k*128 + i], n in [0,64).
// One wave per (d, i-tile); 4 accumulator tiles cover all 64 modes so the
// B fragment (the x data, the HBM-bound stream) is loaded once per K-step.
// v_wmma_f32_16x16x4_f32 fragment layout (ISA 7.12.2, wave32):
//   A 16x4 : lanes 0-15 row M=l, VGPRs = {K0,K1}; lanes 16-31 = {K2,K3}
//   B 4x16 : VGPR0 = row K0(lanes 0-15)/K2(16-31); VGPR1 = K1/K3; N = l
//   C/D    : VGPR r -> row (r, lanes 0-15) / (r+8, lanes 16-31), N = l
// ---------------------------------------------------------------------------
__global__ void dht_gemm(const float* __restrict__ cas1,
                         const float* __restrict__ x,
                         float* __restrict__ H) {
    const int lane = threadIdx.x & 31;
    const int half = lane >> 4;
    const int l    = lane & 15;
    const int i0   = blockIdx.x * 16;           // channel tile [0,128)
    const int d    = blockIdx.y;                // batch
    const float* __restrict__ Xd = x + (size_t)d * (NN * CIN);

    v8f a0c = {0,0,0,0,0,0,0,0};
    v8f a1c = {0,0,0,0,0,0,0,0};
    v8f a2c = {0,0,0,0,0,0,0,0};
    v8f a3c = {0,0,0,0,0,0,0,0};

    const int koff = 2 * half;
    for (int k = 0; k < NN; k += 4) {
        const int kk = k + koff;
        // B fragment: x viewed as [4096 x 128] row-major
        v2f b;
        b.x = Xd[(size_t)kk * CIN + i0 + l];
        b.y = Xd[(size_t)(kk + 1) * CIN + i0 + l];
        // A fragments: 4 mode tiles, contiguous float2 along k
        const float* casr = cas1 + kk;
        v2f a0 = *(const v2f*)(casr + (size_t)(l)      * NN);
        v2f a1 = *(const v2f*)(casr + (size_t)(16 + l) * NN);
        v2f a2 = *(const v2f*)(casr + (size_t)(32 + l) * NN);
        v2f a3 = *(const v2f*)(casr + (size_t)(48 + l) * NN);
        a0c = wmma4(a0, b, a0c);
        a1c = wmma4(a1, b, a1c);
        a2c = wmma4(a2, b, a2c);
        a3c = wmma4(a3, b, a3c);
    }

    float* __restrict__ Hd = H + (size_t)d * (MODES * CIN);
    const int i = i0 + l;
#pragma unroll
    for (int r = 0; r < 8; ++r) {
        const int rr = r + 8 * half;
        Hd[(size_t)(rr)      * CIN + i] = a0c[r];
        Hd[(size_t)(16 + rr) * CIN + i] = a1c[r];
        Hd[(size_t)(32 + rr) * CIN + i] = a2c[r];
        Hd[(size_t)(48 + rr) * CIN + i] = a3c[r];
    }
}

// ---------------------------------------------------------------------------
// Stage B: OM[b][o][nm] = sum_i H[b][nm][i] * W[i][o][nm]
// 64 independent 32x128x128 GEMMs; one wave per (mode, b-tile, o-tile).
// ---------------------------------------------------------------------------
__global__ void mix_gemm(const float* __restrict__ H,
                         const float* __restrict__ W,
                         float* __restrict__ OM) {
    const int lane = threadIdx.x & 31;
    const int half = lane >> 4;
    const int l    = lane & 15;
    const int o0   = blockIdx.x * 16;
    const int b0   = blockIdx.y * 16;
    const int nm   = blockIdx.z;

    v8f acc = {0,0,0,0,0,0,0,0};
    const int koff = 2 * half;
    for (int k = 0; k < CIN; k += 4) {
        const int kk = k + koff;
        v2f a = *(const v2f*)(H + (size_t)(b0 + l) * (MODES * CIN)
                                + (size_t)nm * CIN + kk);
        v2f b;
        b.x = W[(size_t)kk * (COUT * MODES) + (size_t)(o0 + l) * MODES + nm];
        b.y = W[(size_t)(kk + 1) * (COUT * MODES) + (size_t)(o0 + l) * MODES + nm];
        acc = wmma4(a, b, acc);
    }
#pragma unroll
    for (int r = 0; r < 8; ++r) {
        const int bb = b0 + r + 8 * half;
        const int oo = o0 + l;
        OM[(size_t)bb * (COUT * MODES) + (size_t)oo * MODES + nm] = acc[r];
    }
}

// ---------------------------------------------------------------------------
// Stage C: truncated inverse DHT + atan2.
// For output channel m, exactly 64 terms survive the raw-view sparsity:
//   n' in [0,64), m' = (m - n') mod 128,
//   column = m'            -> cas2 col 16*m'      (no wrap, m' <= m)
//   column = 128 + m' - 65 -> cas2 col 16*m' + 1  (wrap,    m' >  m)
//   xr[d,m,n] = (1/2049) * sum_j coeff[j] * T[n][col_j]
//   out[d,m,n] = atan2(xr[d,m,2048-n], xr[d,m,n])
// One 256-thread block per (d, m).
// ---------------------------------------------------------------------------
__global__ void idht_atan(const float* __restrict__ OM,
                          const float* __restrict__ T,
                          float* __restrict__ out) {
    __shared__ float sc[MODES];
    __shared__ int   scol[MODES];
    __shared__ float xr[NHALF];

    const int tid = threadIdx.x;
    const int d   = blockIdx.x >> 7;
    const int m   = blockIdx.x & 127;

    if (tid < MODES) {
        const int np = tid;
        int mp = m - np;
        if (mp < 0) mp += 128;
        const bool wrap = (mp > m);
        scol[tid] = wrap ? (128 + mp - 65) : mp;
        sc[tid]   = OM[(size_t)d * (COUT * MODES) + (size_t)mp * MODES + np];
    }
    __syncthreads();

    const float inv = 1.0f / (float)NHALF;
    for (int n = tid; n < NHALF; n += 256) {
        const float* __restrict__ Tr = T + (size_t)n * TCOLS;
        float s = 0.0f;
#pragma unroll 8
        for (int j = 0; j < MODES; ++j) s += sc[j] * Tr[scol[j]];
        xr[n] = s * inv;
    }
    __syncthreads();

    float* __restrict__ o = out + (size_t)(d * COUT + m) * NHALF;
    for (int n = tid; n < NHALF; n += 256) {
        o[n] = atan2f(xr[(NHALF - 1) - n], xr[n]);
    }
}

// ---------------------------------------------------------------------------
// Launch
//   ws layout (floats): cas1[64*4096] | T[2049*192] | H[32*64*128] | OM[32*128*64]
//   total ~4.7 MB
// ---------------------------------------------------------------------------
extern "C" void kernel_launch(void* const* d_in, const int* in_sizes, int n_in,
                              void* d_out, int out_size, void* d_ws, size_t ws_size,
                              hipStream_t stream) {
    const float* x = (const float*)d_in[0];   // [32,128,4096] f32
    const float* w = (const float*)d_in[1];   // [128,128,64]  f32
    float* out = (float*)d_out;               // [32,128,2049] f32
    float* ws  = (float*)d_ws;

    float* cas1 = ws;                                   // 262144 floats
    float* T    = cas1 + MODES * NN;                    // 393408 floats
    float* H    = T + NHALF * TCOLS;                    // 262144 floats
    float* OM   = H + BB * MODES * CIN;                 // 262144 floats

    build_cas1<<<(MODES * NN + 255) / 256, 256, 0, stream>>>(cas1);
    build_T<<<(NHALF * TCOLS + 255) / 256, 256, 0, stream>>>(T);
    dht_gemm<<<dim3(CIN / 16, BB), 32, 0, stream>>>(cas1, x, H);
    mix_gemm<<<dim3(COUT / 16, BB / 16, MODES), 32, 0, stream>>>(H, w, OM);
    idht_atan<<<BB * COUT, 256, 0, stream>>>(OM, T, out);
}